// GCN_19567871000830
// MI455X (gfx1250) — compile-verified
//
#include <hip/hip_runtime.h>
#include <math.h>

#define BB 8
#define TT 512
#define JJ 24
#define CC 128
#define NB (BB*JJ)       // 192 batches (B*J)
#define KST (CC/4)       // 32 K-steps of 4 for fp32 WMMA
#define STILES (TT/16)   // 32 column tiles
#define NTOT (NB*TT)     // 98304 rows

typedef __attribute__((ext_vector_type(2))) float v2f;
typedef __attribute__((ext_vector_type(8))) float v8f;

__device__ __forceinline__ v8f wmma4(v2f a, v2f b, v8f c) {
  // D = A(16x4) * B(4x16) + C, fp32
  return __builtin_amdgcn_wmma_f32_16x16x4_f32(false, a, false, b, (short)0, c, false, false);
}

#define NEG_BIG (-3.402823466e38f)

// ---------------------------------------------------------------------------
// Kernel 1: vx = xr @ Vw^T + Vb ; ux = xr @ Uw^T + Ub   (one wave = 16 rows)
// ---------------------------------------------------------------------------
__global__ void k_gemm_vu(const float* __restrict__ x,
                          const float* __restrict__ Uw, const float* __restrict__ Ub,
                          const float* __restrict__ Vw, const float* __restrict__ Vb,
                          float* __restrict__ vx, float* __restrict__ ux) {
  int tile = blockIdx.x;
  int n  = tile >> 5;
  int t0 = (tile & 31) << 4;
  int b = n / JJ, j = n % JJ;
  int lane = threadIdx.x & 31;
  int hi = lane >> 4, lm = lane & 15;

  // A fragments: lane holds row t0+lm, K pair (2*hi, 2*hi+1) per K-step
  const float* arow = x + (((size_t)b*TT + t0 + lm)*JJ + j)*CC + 2*hi;
  v2f a[KST];
#pragma unroll
  for (int k = 0; k < KST; ++k) a[k] = *(const v2f*)(arow + 4*k);

  for (int nt = 0; nt < 8; ++nt) {
    v8f cv = {}; v8f cu = {};
    const float* vrow = Vw + (size_t)(nt*16 + lm)*CC + 2*hi;  // B[k,n] = Vw[n,k]
    const float* urow = Uw + (size_t)(nt*16 + lm)*CC + 2*hi;
#pragma unroll
    for (int k = 0; k < KST; ++k) {
      v2f bv = *(const v2f*)(vrow + 4*k);
      cv = wmma4(a[k], bv, cv);
      v2f bu = *(const v2f*)(urow + 4*k);
      cu = wmma4(a[k], bu, cu);
    }
#pragma unroll
    for (int r = 0; r < 8; ++r) {
      int row = t0 + r + 8*hi;
      int col = nt*16 + lm;
      size_t o = (size_t)(n*TT + row)*CC + col;
      vx[o] = cv[r] + Vb[col];
      ux[o] = cu[r] + Ub[col];
    }
  }
}

// ---------------------------------------------------------------------------
// Kernel 2: sim strip (WMMA) -> per-row top-4 threshold + degree^{-1/2}
// ---------------------------------------------------------------------------
__global__ void k_thr_deg(const float* __restrict__ x,
                          float* __restrict__ thr, float* __restrict__ dis) {
  __shared__ float ssim[16*TT];  // 32 KB strip: 16 rows x 512 cols
  int tile = blockIdx.x;
  int n  = tile >> 5;
  int t0 = (tile & 31) << 4;
  int b = n / JJ, j = n % JJ;
  int lane = threadIdx.x & 31;
  int hi = lane >> 4, lm = lane & 15;

  const float* arow = x + (((size_t)b*TT + t0 + lm)*JJ + j)*CC + 2*hi;
  v2f a[KST];
#pragma unroll
  for (int k = 0; k < KST; ++k) a[k] = *(const v2f*)(arow + 4*k);

  float tp0[8], tp1[8], tp2[8], tp3[8];   // descending top-4 per row (per lane)
#pragma unroll
  for (int r = 0; r < 8; ++r) { tp0[r]=tp1[r]=tp2[r]=tp3[r]=NEG_BIG; }

  for (int st = 0; st < STILES; ++st) {
    const float* brow = x + (((size_t)b*TT + st*16 + lm)*JJ + j)*CC + 2*hi;
    v8f c = {};
#pragma unroll
    for (int k = 0; k < KST; ++k) {
      v2f bf = *(const v2f*)(brow + 4*k);
      c = wmma4(a[k], bf, c);
    }
#pragma unroll
    for (int r = 0; r < 8; ++r) {
      float v = c[r];
      ssim[(r + 8*hi)*TT + st*16 + lm] = v;
      float y = v, m;
      m = fmaxf(tp0[r], y); y = fminf(tp0[r], y); tp0[r] = m;
      m = fmaxf(tp1[r], y); y = fminf(tp1[r], y); tp1[r] = m;
      m = fmaxf(tp2[r], y); y = fminf(tp2[r], y); tp2[r] = m;
      tp3[r] = fmaxf(tp3[r], y);
    }
  }

  // merge top-4 lists across the 16 lanes of each half-wave
#pragma unroll
  for (int msk = 1; msk <= 8; msk <<= 1) {
#pragma unroll
    for (int r = 0; r < 8; ++r) {
      float o0 = __shfl_xor(tp0[r], msk, 32);
      float o1 = __shfl_xor(tp1[r], msk, 32);
      float o2 = __shfl_xor(tp2[r], msk, 32);
      float o3 = __shfl_xor(tp3[r], msk, 32);
      float vals[4] = {o0, o1, o2, o3};
#pragma unroll
      for (int i = 0; i < 4; ++i) {
        float y = vals[i], mm;
        mm = fmaxf(tp0[r], y); y = fminf(tp0[r], y); tp0[r] = mm;
        mm = fmaxf(tp1[r], y); y = fminf(tp1[r], y); tp1[r] = mm;
        mm = fmaxf(tp2[r], y); y = fminf(tp2[r], y); tp2[r] = mm;
        tp3[r] = fmaxf(tp3[r], y);
      }
    }
  }

  // degree = #(sim >= threshold) over 512 cols, from LDS strip
#pragma unroll
  for (int r = 0; r < 8; ++r) {
    float th = tp3[r];
    float d = 0.f;
    for (int st = 0; st < STILES; ++st)
      d += (ssim[(r + 8*hi)*TT + st*16 + lm] >= th) ? 1.f : 0.f;
    d += __shfl_xor(d, 1, 32);
    d += __shfl_xor(d, 2, 32);
    d += __shfl_xor(d, 4, 32);
    d += __shfl_xor(d, 8, 32);
    if (lm == 0) {
      int row = t0 + r + 8*hi;
      thr[n*TT + row] = th;
      dis[n*TT + row] = rsqrtf(d);
    }
  }
}

// ---------------------------------------------------------------------------
// Kernel 3: recompute sim strip, form norm_adj on the fly, out = nA@vx + ux
// Writes pre-BN result into d_out at permuted (B,T,J,C) position; accumulates
// per-channel sum / sumsq for batch-norm via global fp32 atomics.
// ---------------------------------------------------------------------------
__global__ void k_aggregate(const float* __restrict__ x,
                            const float* __restrict__ vx, const float* __restrict__ ux,
                            const float* __restrict__ thr, const float* __restrict__ dis,
                            float* __restrict__ out, float* __restrict__ stat) {
  __shared__ float ssim[16*TT];  // 32 KB
  __shared__ float sdis[TT];     //  2 KB
  int tile = blockIdx.x;
  int n  = tile >> 5;
  int t0 = (tile & 31) << 4;
  int b = n / JJ, j = n % JJ;
  int lane = threadIdx.x & 31;
  int hi = lane >> 4, lm = lane & 15;

  for (int i = lane; i < TT; i += 32) sdis[i] = dis[n*TT + i];

  const float* arow = x + (((size_t)b*TT + t0 + lm)*JJ + j)*CC + 2*hi;
  v2f a[KST];
#pragma unroll
  for (int k = 0; k < KST; ++k) a[k] = *(const v2f*)(arow + 4*k);

  // pass A: sim strip into LDS
  for (int st = 0; st < STILES; ++st) {
    const float* brow = x + (((size_t)b*TT + st*16 + lm)*JJ + j)*CC + 2*hi;
    v8f c = {};
#pragma unroll
    for (int k = 0; k < KST; ++k) {
      v2f bf = *(const v2f*)(brow + 4*k);
      c = wmma4(a[k], bf, c);
    }
#pragma unroll
    for (int r = 0; r < 8; ++r)
      ssim[(r + 8*hi)*TT + st*16 + lm] = c[r];
  }

  float throw_ = thr[n*TT + t0 + lm];   // A-layout rows: lane -> row lm
  float disr   = dis[n*TT + t0 + lm];

  // init accumulators from ux (C layout: VGPR r -> row r+8*hi, lane -> col)
  v8f co[8];
#pragma unroll
  for (int nt = 0; nt < 8; ++nt) {
    v8f z = {};
#pragma unroll
    for (int r = 0; r < 8; ++r)
      z[r] = ux[(size_t)(n*TT + t0 + r + 8*hi)*CC + nt*16 + lm];
    co[nt] = z;
  }

  // pass B: out += norm_adj @ vx, K = 512 in steps of 4
  for (int ks = 0; ks < TT/4; ++ks) {
    int s0 = 4*ks + 2*hi;
    v2f sm = *(const v2f*)(&ssim[lm*TT + s0]);
    float a0 = (sm.x >= throw_) ? (disr * sdis[s0])     : 0.f;
    float a1 = (sm.y >= throw_) ? (disr * sdis[s0 + 1]) : 0.f;
    v2f af = {a0, a1};
    const float* v0 = vx + (size_t)(n*TT + s0)*CC + lm;
    const float* v1 = v0 + CC;
#pragma unroll
    for (int nt = 0; nt < 8; ++nt) {
      v2f bf = { v0[nt*16], v1[nt*16] };
      co[nt] = wmma4(af, bf, co[nt]);
    }
  }

  // store pre-BN output into permuted (b,t,j,c) layout + BN partial stats
#pragma unroll
  for (int nt = 0; nt < 8; ++nt) {
    float s = 0.f, s2 = 0.f;
#pragma unroll
    for (int r = 0; r < 8; ++r) {
      float v = co[nt][r];
      int row = t0 + r + 8*hi;
      int col = nt*16 + lm;
      out[(((size_t)b*TT + row)*JJ + j)*CC + col] = v;
      s += v; s2 += v*v;
    }
    s  += __shfl_xor(s,  16, 32);
    s2 += __shfl_xor(s2, 16, 32);
    if (hi == 0) {
      int col = nt*16 + lm;
      atomicAdd(&stat[col],      s);
      atomicAdd(&stat[CC + col], s2);
    }
  }
}

// ---------------------------------------------------------------------------
// Kernel 4: batch-norm finalize + residual + ReLU (in output layout)
// ---------------------------------------------------------------------------
__global__ void k_final(const float* __restrict__ x, const float* __restrict__ gamma,
                        const float* __restrict__ beta, const float* __restrict__ stat,
                        float* __restrict__ out) {
  size_t i = (size_t)blockIdx.x * blockDim.x + threadIdx.x;
  size_t total = (size_t)BB*TT*JJ*CC;
  if (i >= total) return;
  int c = (int)(i & (CC - 1));
  float inv = 1.0f / (float)NTOT;
  float mean = stat[c] * inv;
  float var  = stat[CC + c] * inv - mean * mean;
  float scale = rsqrtf(var + 1e-5f) * gamma[c];
  float v = out[i];
  float bn = (v - mean) * scale + beta[c];
  out[i] = fmaxf(x[i] + bn, 0.f);
}

__global__ void k_zero(float* __restrict__ stat) {
  if (threadIdx.x < 2*CC) stat[threadIdx.x] = 0.f;
}

// ---------------------------------------------------------------------------
extern "C" void kernel_launch(void* const* d_in, const int* in_sizes, int n_in,
                              void* d_out, int out_size, void* d_ws, size_t ws_size,
                              hipStream_t stream) {
  const float* x     = (const float*)d_in[0];
  const float* Uw    = (const float*)d_in[1];
  const float* Ub    = (const float*)d_in[2];
  const float* Vw    = (const float*)d_in[3];
  const float* Vb    = (const float*)d_in[4];
  const float* gamma = (const float*)d_in[5];
  const float* beta  = (const float*)d_in[6];
  float* out = (float*)d_out;

  float* ws  = (float*)d_ws;
  float* vx  = ws;                               // 12,582,912 f32
  float* ux  = vx  + (size_t)NB*TT*CC;           // 12,582,912 f32
  float* thr = ux  + (size_t)NB*TT*CC;           //     98,304 f32
  float* dis = thr + (size_t)NB*TT;              //     98,304 f32
  float* stat = dis + (size_t)NB*TT;             //        256 f32

  const int tiles = NTOT / 16;                   // 6144 row-tiles

  k_zero<<<1, 256, 0, stream>>>(stat);
  k_gemm_vu<<<tiles, 32, 0, stream>>>(x, Uw, Ub, Vw, Vb, vx, ux);
  k_thr_deg<<<tiles, 32, 0, stream>>>(x, thr, dis);
  k_aggregate<<<tiles, 32, 0, stream>>>(x, vx, ux, thr, dis, out, stat);

  size_t total = (size_t)BB*TT*JJ*CC;
  k_final<<<(int)((total + 255) / 256), 256, 0, stream>>>(x, gamma, beta, stat, out);
}